// MaskingMixture_88691074662523
// MI455X (gfx1250) — compile-verified
//
#include <hip/hip_runtime.h>

#define BATCH 32
#define NTOK  524288          // 16*32*32*32
#define EPSC  1e-3f
#define NBIN  65536

typedef float v8f __attribute__((ext_vector_type(8)));
typedef float v2f __attribute__((ext_vector_type(2)));

__device__ __forceinline__ float clampf(float x) {
    return fminf(fmaxf(x, EPSC), 1.0f - EPSC);
}

// order-preserving float -> uint32 (ascending); top-K == largest keys
__device__ __forceinline__ unsigned keyOf(float p) {
    unsigned b = __float_as_uint(p);
    return (b & 0x80000000u) ? ~b : (b | 0x80000000u);
}

// Build per-row LDS tables: lvlt[0..15]=log(u_v)/a0, lvlt[16..47]=log(u_t)/a1,
// hw[32*32] = log(u_h[h])/a2 + log(u_w[w])/a3 via f32 WMMA outer-product sums.
__device__ __forceinline__ void build_tables(const float* __restrict__ dl,
                                             float* hw, float* lvlt) {
    int tid = threadIdx.x;
    if (tid < 48) lvlt[tid] = dl[tid];
    if (tid < 32) {                       // wave 0, EXEC all ones
        int lane = tid;
        const float* lh = dl + 48;
        const float* lw = dl + 80;
        bool lo = lane < 16;
        #pragma unroll
        for (int hi = 0; hi < 2; ++hi) {
            #pragma unroll
            for (int wi = 0; wi < 2; ++wi) {
                v2f a, b;
                // A (16x4): col0 = lh chunk, col1 = ones, cols 2,3 = 0
                a.x = lo ? lh[hi * 16 + (lane & 15)] : 0.0f;   // K=0 / K=2
                a.y = lo ? 1.0f : 0.0f;                        // K=1 / K=3
                // B (4x16): row0 = ones, row1 = lw chunk, rows 2,3 = 0
                b.x = lo ? 1.0f : 0.0f;
                b.y = lo ? lw[wi * 16 + (lane & 15)] : 0.0f;
                v8f c = {};
                c = __builtin_amdgcn_wmma_f32_16x16x4_f32(
                        false, a, false, b, (short)0, c, false, false);
                int mb = hi * 16 + (lo ? 0 : 8);
                int nn = wi * 16 + (lane & 15);
                #pragma unroll
                for (int r = 0; r < 8; ++r)
                    hw[(mb + r) * 32 + nn] = c[r];
            }
        }
    }
}

__device__ __forceinline__ void keys4(float4 u, int n0, const float* hw,
                                      float vt, unsigned* k) {
    k[0] = keyOf(__logf(clampf(u.x)) + vt + hw[(n0 + 0) & 1023]);
    k[1] = keyOf(__logf(clampf(u.y)) + vt + hw[(n0 + 1) & 1023]);
    k[2] = keyOf(__logf(clampf(u.z)) + vt + hw[(n0 + 2) & 1023]);
    k[3] = keyOf(__logf(clampf(u.w)) + vt + hw[(n0 + 3) & 1023]);
}

__global__ __launch_bounds__(256) void zero_kernel(uint4* p, size_t n4) {
    size_t i = (size_t)blockIdx.x * blockDim.x + threadIdx.x;
    size_t stride = (size_t)gridDim.x * blockDim.x;
    uint4 z = make_uint4(0u, 0u, 0u, 0u);
    for (; i < n4; i += stride) p[i] = z;
}

__global__ __launch_bounds__(256) void precompute_kernel(
        const float* __restrict__ components, const int* __restrict__ idxPtr,
        const float* __restrict__ urs, const float* __restrict__ urt,
        const float* __restrict__ uv_s, const float* __restrict__ ut_s,
        const float* __restrict__ uh_s, const float* __restrict__ uw_s,
        const float* __restrict__ uv_t, const float* __restrict__ ut_t,
        const float* __restrict__ uh_t, const float* __restrict__ uw_t,
        float* __restrict__ dimLog, int* __restrict__ Kbuf) {
    int idx = idxPtr[0];
    float a0 = components[idx * 4 + 0];
    float a1 = components[idx * 4 + 1];
    float a2 = components[idx * 4 + 2];
    float a3 = components[idx * 4 + 3];
    int tid = threadIdx.x;
    if (tid == 0) {
        Kbuf[0] = (int)(clampf(urs[0]) * (float)NTOK);
        Kbuf[1] = (int)(clampf(urt[0]) * (float)NTOK);
    }
    for (int e = tid; e < 2 * BATCH * 112; e += blockDim.x) {
        int s = e / (BATCH * 112);
        int rem = e % (BATCH * 112);
        int b = rem / 112;
        int k = rem % 112;
        const float* src; int off; float a;
        if (k < 16)      { a = a0; src = s ? uv_t : uv_s; off = b * 16 + k; }
        else if (k < 48) { a = a1; src = s ? ut_t : ut_s; off = b * 32 + (k - 16); }
        else if (k < 80) { a = a2; src = s ? uh_t : uh_s; off = b * 32 + (k - 48); }
        else             { a = a3; src = s ? uw_t : uw_s; off = b * 32 + (k - 80); }
        dimLog[e] = __logf(clampf(src[off])) / a;
    }
}

__global__ __launch_bounds__(256) void hist_pass(
        const float* __restrict__ u0s, const float* __restrict__ u0t,
        const float* __restrict__ dimLog, unsigned* __restrict__ hist) {
    const int side = blockIdx.z, row = blockIdx.y, tid = threadIdx.x;
    __shared__ float hw[1024];
    __shared__ float lvlt[48];
    const int id = side * BATCH + row;
    build_tables(dimLog + id * 112, hw, lvlt);
    __syncthreads();
    const float* u0 = (side ? u0t : u0s) + (size_t)row * NTOK;
    unsigned* h = hist + (size_t)id * NBIN;
    const int base = blockIdx.x * 2048;
    #pragma unroll
    for (int half = 0; half < 2; ++half) {
        int n0 = base + half * 1024 + tid * 4;
        float4 u = *(const float4*)(u0 + n0);
        float vt = lvlt[n0 >> 15] + lvlt[16 + ((n0 >> 10) & 31)];
        unsigned k[4];
        keys4(u, n0, hw, vt, k);
        atomicAdd(&h[k[0] >> 16], 1u);
        atomicAdd(&h[k[1] >> 16], 1u);
        atomicAdd(&h[k[2] >> 16], 1u);
        atomicAdd(&h[k[3] >> 16], 1u);
    }
}

__global__ __launch_bounds__(256) void scan1_kernel(
        const unsigned* __restrict__ hist, const int* __restrict__ Kbuf,
        int* __restrict__ threshBin, int* __restrict__ countAbove) {
    int row = blockIdx.x, side = blockIdx.y, tid = threadIdx.x;
    int id = side * BATCH + row;
    const unsigned* h = hist + (size_t)id * NBIN;
    __shared__ unsigned partial[256];
    unsigned s = 0;
    for (int i = 0; i < 256; ++i) s += h[tid * 256 + i];
    partial[tid] = s;
    __syncthreads();
    if (tid == 0) {
        unsigned K = (unsigned)Kbuf[side];
        unsigned acc = 0, before = 0;
        int bin = 0;
        for (int c = 255; c >= 0; --c) {
            if (acc + partial[c] >= K) {
                for (int b2 = c * 256 + 255; b2 >= c * 256; --b2) {
                    unsigned cnt = h[b2];
                    if (acc + cnt >= K) { bin = b2; before = acc; break; }
                    acc += cnt;
                }
                break;
            }
            acc += partial[c];
        }
        threshBin[id] = bin;
        countAbove[id] = (int)before;
    }
}

__global__ __launch_bounds__(256) void hist2_pass(
        const float* __restrict__ u0s, const float* __restrict__ u0t,
        const float* __restrict__ dimLog, const int* __restrict__ threshBin,
        unsigned* __restrict__ hist2) {
    const int side = blockIdx.z, row = blockIdx.y, tid = threadIdx.x;
    __shared__ float hw[1024];
    __shared__ float lvlt[48];
    const int id = side * BATCH + row;
    build_tables(dimLog + id * 112, hw, lvlt);
    __syncthreads();
    const float* u0 = (side ? u0t : u0s) + (size_t)row * NTOK;
    unsigned* h2 = hist2 + (size_t)id * NBIN;
    const unsigned tb = (unsigned)threshBin[id];
    const int base = blockIdx.x * 2048;
    #pragma unroll
    for (int half = 0; half < 2; ++half) {
        int n0 = base + half * 1024 + tid * 4;
        float4 u = *(const float4*)(u0 + n0);
        float vt = lvlt[n0 >> 15] + lvlt[16 + ((n0 >> 10) & 31)];
        unsigned k[4];
        keys4(u, n0, hw, vt, k);
        #pragma unroll
        for (int j = 0; j < 4; ++j)
            if ((k[j] >> 16) == tb) atomicAdd(&h2[k[j] & 0xFFFFu], 1u);
    }
}

__global__ __launch_bounds__(256) void scan2_kernel(
        const unsigned* __restrict__ hist2, const int* __restrict__ Kbuf,
        const int* __restrict__ threshBin, const int* __restrict__ countAbove,
        unsigned* __restrict__ threshKey, int* __restrict__ tieAllow,
        unsigned* __restrict__ ticket) {
    int row = blockIdx.x, side = blockIdx.y, tid = threadIdx.x;
    int id = side * BATCH + row;
    const unsigned* h = hist2 + (size_t)id * NBIN;
    __shared__ unsigned partial[256];
    unsigned s = 0;
    for (int i = 0; i < 256; ++i) s += h[tid * 256 + i];
    partial[tid] = s;
    __syncthreads();
    if (tid == 0) {
        unsigned target = (unsigned)(Kbuf[side] - countAbove[id]);
        unsigned acc = 0, before = 0;
        int bin = 0;
        for (int c = 255; c >= 0; --c) {
            if (acc + partial[c] >= target) {
                for (int b2 = c * 256 + 255; b2 >= c * 256; --b2) {
                    unsigned cnt = h[b2];
                    if (acc + cnt >= target) { bin = b2; before = acc; break; }
                    acc += cnt;
                }
                break;
            }
            acc += partial[c];
        }
        threshKey[id] = ((unsigned)threshBin[id] << 16) | (unsigned)bin;
        tieAllow[id] = (int)(target - before);
        ticket[id] = 0u;
    }
}

__global__ __launch_bounds__(256) void mask_pass(
        const float* __restrict__ u0s, const float* __restrict__ u0t,
        const float* __restrict__ dimLog, const unsigned* __restrict__ threshKey,
        const int* __restrict__ tieAllow, unsigned* __restrict__ ticket,
        float* __restrict__ out) {
    const int side = blockIdx.z, row = blockIdx.y, tid = threadIdx.x;
    __shared__ float hw[1024];
    __shared__ float lvlt[48];
    const int id = side * BATCH + row;
    build_tables(dimLog + id * 112, hw, lvlt);
    __syncthreads();
    const float* u0 = (side ? u0t : u0s) + (size_t)row * NTOK;
    float* orow = out + (size_t)id * NTOK;
    const unsigned tk = threshKey[id];
    const unsigned ta = (unsigned)tieAllow[id];
    unsigned* tick = &ticket[id];
    const int base = blockIdx.x * 2048;
    #pragma unroll
    for (int half = 0; half < 2; ++half) {
        int n0 = base + half * 1024 + tid * 4;
        float4 u = *(const float4*)(u0 + n0);
        float vt = lvlt[n0 >> 15] + lvlt[16 + ((n0 >> 10) & 31)];
        unsigned k[4];
        keys4(u, n0, hw, vt, k);
        float v[4];
        #pragma unroll
        for (int j = 0; j < 4; ++j) {
            if (k[j] > tk)       v[j] = 1.0f;
            else if (k[j] == tk) v[j] = (atomicAdd(tick, 1u) < ta) ? 1.0f : 0.0f;
            else                 v[j] = 0.0f;
        }
        *(float4*)(orow + n0) = make_float4(v[0], v[1], v[2], v[3]);
    }
}

extern "C" void kernel_launch(void* const* d_in, const int* in_sizes, int n_in,
                              void* d_out, int out_size, void* d_ws, size_t ws_size,
                              hipStream_t stream) {
    (void)in_sizes; (void)n_in; (void)out_size; (void)ws_size;
    const float* components = (const float*)d_in[0];
    const float* u_rate_src = (const float*)d_in[1];
    const float* u_rate_tgt = (const float*)d_in[2];
    const float* u0_src     = (const float*)d_in[3];
    const float* u0_tgt     = (const float*)d_in[4];
    const float* uv_s = (const float*)d_in[5];
    const float* ut_s = (const float*)d_in[6];
    const float* uh_s = (const float*)d_in[7];
    const float* uw_s = (const float*)d_in[8];
    const float* uv_t = (const float*)d_in[9];
    const float* ut_t = (const float*)d_in[10];
    const float* uh_t = (const float*)d_in[11];
    const float* uw_t = (const float*)d_in[12];
    const int*   idx  = (const int*)d_in[13];
    float* out = (float*)d_out;

    char* ws = (char*)d_ws;
    const size_t HISTB = (size_t)2 * BATCH * NBIN * sizeof(unsigned); // 16 MB
    size_t off = 0;
    unsigned* hist1 = (unsigned*)(ws + off); off += HISTB;
    unsigned* hist2 = (unsigned*)(ws + off); off += HISTB;
    float*    dimLog = (float*)(ws + off);   off += (size_t)2 * BATCH * 112 * sizeof(float);
    int*      Kbuf   = (int*)(ws + off);     off += 2 * sizeof(int);
    int*      tBin   = (int*)(ws + off);     off += 2 * BATCH * sizeof(int);
    int*      cAbove = (int*)(ws + off);     off += 2 * BATCH * sizeof(int);
    unsigned* tKey   = (unsigned*)(ws + off); off += 2 * BATCH * sizeof(unsigned);
    int*      tAllow = (int*)(ws + off);     off += 2 * BATCH * sizeof(int);
    unsigned* ticket = (unsigned*)(ws + off); off += 2 * BATCH * sizeof(unsigned);

    // 1) zero both histogram regions (32 MB) each call for determinism
    zero_kernel<<<2048, 256, 0, stream>>>((uint4*)ws, (2 * HISTB) / 16);

    // 2) tiny precompute: alphas, K, per-row log tables
    precompute_kernel<<<1, 256, 0, stream>>>(components, idx, u_rate_src, u_rate_tgt,
                                             uv_s, ut_s, uh_s, uw_s,
                                             uv_t, ut_t, uh_t, uw_t,
                                             dimLog, Kbuf);

    dim3 big(NTOK / 2048, BATCH, 2);
    dim3 small(BATCH, 2);

    // 3) radix-select pass 1: high-16-bit histogram
    hist_pass<<<big, 256, 0, stream>>>(u0_src, u0_tgt, dimLog, hist1);
    scan1_kernel<<<small, 256, 0, stream>>>(hist1, Kbuf, tBin, cAbove);

    // 4) radix-select pass 2: low-16-bit refinement within threshold bucket
    hist2_pass<<<big, 256, 0, stream>>>(u0_src, u0_tgt, dimLog, tBin, hist2);
    scan2_kernel<<<small, 256, 0, stream>>>(hist2, Kbuf, tBin, cAbove, tKey, tAllow, ticket);

    // 5) final mask write
    mask_pass<<<big, 256, 0, stream>>>(u0_src, u0_tgt, dimLog, tKey, tAllow, ticket, out);
}